// Net_pool_84696755077755
// MI455X (gfx1250) — compile-verified
//
#include <hip/hip_runtime.h>

typedef __attribute__((ext_vector_type(2))) float v2f;
typedef __attribute__((ext_vector_type(8))) float v8f;
typedef __attribute__((ext_vector_type(4))) unsigned int u32x4;
typedef __attribute__((ext_vector_type(8))) int i32x8;
typedef __attribute__((ext_vector_type(4))) int i32x4;

// ---- problem constants ----
#define NFEAT      169          // in/out features
#define GRID_E     14           // 14x14 padded grid
#define POOLED     36           // 6x6 pooled
#define HID        128
#define NTILE2     11           // ceil(169/16)

// ---- block shape ----
#define WAVES          8
#define ROWS_PER_WAVE  16
#define ROWS_PER_BLOCK (WAVES * ROWS_PER_WAVE)   // 128

// ---- dynamic LDS layout (float offsets) ----
#define W1_STRIDE  38                       // even stride -> b64-aligned (k,k+1) pairs
#define W1_OFF     0                        // 128*38      = 4864 floats
#define W2_OFF     4864                     // 176*128     = 22528 floats
#define B1_OFF     27392                    // 128
#define B2_OFF     27520                    // 176 (padded, zeros past 169)
#define OVL_OFF    27696                    // per-wave overlay: xs(16x196) then h1(16x128)
#define XS_STRIDE  196
#define OVL_PER_WAVE (ROWS_PER_WAVE * XS_STRIDE)      // 3136 floats
#define SMEM_FLOATS (OVL_OFF + WAVES * OVL_PER_WAVE)  // 52784
#define SMEM_BYTES  (SMEM_FLOATS * 4)                 // 211136 B (<320KB/WG on CDNA5)

__device__ __forceinline__ unsigned rfl(unsigned v) {
    return (unsigned)__builtin_amdgcn_readfirstlane((int)v);
}

// TDM iterate-mode load: nrows tiles of (row_elems x 1) f32, per-iteration
// increments g_stride_elems (global) / lds_stride_elems (LDS), both in elements.
// Handles strided re-layout (e.g. 169 -> 196) fully in the DMA engine.
__device__ __forceinline__ void tdm_load_rows(
    const float* gsrc, unsigned lds_byte, unsigned row_elems,
    unsigned nrows, unsigned g_stride_elems, unsigned lds_stride_elems)
{
    unsigned long long ga = (unsigned long long)(uintptr_t)gsrc;
    unsigned galo = rfl((unsigned)ga);
    unsigned gahi = rfl((unsigned)(ga >> 32));
    unsigned lb   = rfl(lds_byte);

    u32x4 g0;
    g0[0] = 1u;                                   // count=1 (valid), user mode
    g0[1] = lb;                                   // lds_addr (bytes)
    g0[2] = galo;                                 // global_addr[31:0]
    g0[3] = (gahi & 0x01ffffffu) | (2u << 30);    // global_addr[56:32] | type=2

    i32x8 g1;
    g1[0] = (2 << 16) | (1 << 19);                // data_size=4B, iterate_enable=1
    g1[1] = (int)(row_elems << 16);               // tensor_dim0[15:0]
    g1[2] = (int)(1u << 16);                      // tensor_dim0[31:16]=0 | tensor_dim1=1
    g1[3] = (int)(row_elems << 16);               // tensor_dim1 hi=0 | tile_dim0
    g1[4] = 1;                                    // tile_dim1=1, tile_dim2=0
    g1[5] = (int)row_elems;                       // tensor_dim0_stride[31:0]
    g1[6] = 0;                                    // stride0 hi | dim1_stride lo
    g1[7] = 0;                                    // dim1_stride hi

    i32x4 g2;
    g2[0] = 0;                                    // tensor_dim2 (unused)
    g2[1] = (int)lds_stride_elems;                // lds_addr_increment (elements)
    g2[2] = (int)g_stride_elems;                  // global_addr_increment[31:0]
    g2[3] = (int)((nrows - 1u) << 16);            // gai hi=0 | iterate_count

    i32x4 g3 = {0, 0, 0, 0};

#if defined(__clang_major__) && (__clang_major__ >= 23)
    i32x8 g4 = {0, 0, 0, 0, 0, 0, 0, 0};
    __builtin_amdgcn_tensor_load_to_lds(g0, g1, g2, g3, g4, 0);
#else
    __builtin_amdgcn_tensor_load_to_lds(g0, g1, g2, g3, 0);
#endif
}

__global__ __launch_bounds__(256) void net_pool_mlp_kernel(
    const float* __restrict__ x,  const float* __restrict__ w1,
    const float* __restrict__ b1, const float* __restrict__ w2,
    const float* __restrict__ b2, float* __restrict__ out)
{
    extern __shared__ float smem[];
    const int t    = threadIdx.x;
    const int wave = t >> 5;
    const int lane = t & 31;
    const int lo   = lane & 15;   // M row (A) or N col (B/C)
    const int hi   = lane >> 4;   // K-pair selector

    float* w1s = smem + W1_OFF;
    float* w2s = smem + W2_OFF;
    float* b1s = smem + B1_OFF;
    float* b2s = smem + B2_OFF;

    const int row0 = blockIdx.x * ROWS_PER_BLOCK + wave * ROWS_PER_WAVE;
    float* xs = smem + OVL_OFF + wave * OVL_PER_WAVE;        // [16][196]
    const float* xsrc = x + (size_t)row0 * NFEAT;

    // ---------------- issue all TDM DMA transfers ----------------
    if (wave == 0) {                          // w2: 169x128 flat (86.5 KB), one shot
        tdm_load_rows(w2, (unsigned)(uintptr_t)w2s, NFEAT * HID, 1, 0, 0);
    }
    if (wave == 1) {                          // w1: 128 rows of 36, re-strided to 38
        tdm_load_rows(w1, (unsigned)(uintptr_t)w1s, POOLED, HID, POOLED, W1_STRIDE);
    }
    // per-wave x tile: 16 rows of 169, re-strided to 196 (padded pooling grid)
    tdm_load_rows(xsrc, (unsigned)(uintptr_t)xs, NFEAT, ROWS_PER_WAVE, NFEAT, XS_STRIDE);

    // ---------------- small staging done by ALU while DMA runs ----------------
    #pragma unroll
    for (int it = 0; it < 4; ++it) {              // zero-pad w2 rows 169..175
        int i = it * 256 + t;
        if (i < 7 * HID) w2s[NFEAT * HID + i] = 0.0f;
    }
    if (t < HID) b1s[t] = b1[t];
    if (t < 176) b2s[t] = (t < NFEAT) ? b2[t] : 0.0f;
    #pragma unroll
    for (int it = 0; it < 14; ++it) {             // zero xs pad cols 169..195
        int idx = it * 32 + lane;
        if (idx < 16 * 27) {
            int r = idx / 27;
            int c = NFEAT + (idx - r * 27);
            xs[r * XS_STRIDE + c] = 0.0f;
        }
    }

    __builtin_amdgcn_s_wait_tensorcnt(0);         // all TDM data landed in LDS
    __syncthreads();

    // ---------------- pooling + ReLU, computed directly in A-fragment layout ----------------
    // lane holds h0[M=lo][K = 4*kc + 2*hi + j]  (f32 WMMA 16x16x4 A layout)
    float a[18];
    #pragma unroll
    for (int kc = 0; kc < 9; ++kc) {
        #pragma unroll
        for (int j = 0; j < 2; ++j) {
            int pc = 4 * kc + 2 * hi + j;        // pooled index 0..35
            int p  = pc / 6;
            int q  = pc - 6 * p;
            const float* base = xs + lo * XS_STRIDE + (2 * p) * GRID_E + 2 * q;
            float s = 0.0f;
            #pragma unroll
            for (int i = 0; i < 3; ++i) {
                const float* rp = base + i * GRID_E;
                s += rp[0] + rp[1] + rp[2];
            }
            float h = s * (1.0f / 9.0f);
            a[kc * 2 + j] = h > 0.0f ? h : 0.0f;
        }
    }
    __syncthreads();   // xs reads done everywhere before h1 overlays it

    // ---------------- GEMM1: h0[16x36] x w1^T[36x128] -> h1 (ReLU) into LDS ----------------
    float* h1s = smem + OVL_OFF + wave * OVL_PER_WAVE;       // [16][128], overlays xs
    for (int nt = 0; nt < HID / 16; ++nt) {
        float bias = b1s[nt * 16 + lo];
        v8f acc;
        #pragma unroll
        for (int r = 0; r < 8; ++r) acc[r] = bias;
        #pragma unroll
        for (int kc = 0; kc < 9; ++kc) {
            v2f af; af[0] = a[2 * kc]; af[1] = a[2 * kc + 1];
            v2f bf = *(const v2f*)(w1s + (nt * 16 + lo) * W1_STRIDE + 4 * kc + 2 * hi);
            acc = __builtin_amdgcn_wmma_f32_16x16x4_f32(
                false, af, false, bf, (short)0, acc, false, false);
        }
        #pragma unroll
        for (int r = 0; r < 8; ++r) {
            float v = acc[r] > 0.0f ? acc[r] : 0.0f;
            h1s[(r + 8 * hi) * HID + nt * 16 + lo] = v;      // D layout: M=r+8*hi, N=16*nt+lo
        }
    }
    __syncthreads();

    // ---------------- GEMM2: h1[16x128] x w2^T[128x176] -> out ----------------
    v2f af2[32];                                 // hoist all A-fragments (64 VGPRs)
    #pragma unroll
    for (int kc = 0; kc < 32; ++kc)
        af2[kc] = *(const v2f*)(h1s + lo * HID + 4 * kc + 2 * hi);

    for (int nt = 0; nt < NTILE2; ++nt) {
        int n0 = nt * 16;
        float bias = b2s[n0 + lo];
        v8f acc;
        #pragma unroll
        for (int r = 0; r < 8; ++r) acc[r] = bias;
        #pragma unroll
        for (int kc = 0; kc < 32; ++kc) {
            v2f bf = *(const v2f*)(w2s + (n0 + lo) * HID + 4 * kc + 2 * hi);
            acc = __builtin_amdgcn_wmma_f32_16x16x4_f32(
                false, af2[kc], false, bf, (short)0, acc, false, false);
        }
        int col = n0 + lo;
        if (col < NFEAT) {
            #pragma unroll
            for (int r = 0; r < 8; ++r)
                out[(size_t)(row0 + r + 8 * hi) * NFEAT + col] = acc[r];
        }
    }
}

extern "C" void kernel_launch(void* const* d_in, const int* in_sizes, int n_in,
                              void* d_out, int out_size, void* d_ws, size_t ws_size,
                              hipStream_t stream) {
    const float* x  = (const float*)d_in[0];
    const float* w1 = (const float*)d_in[1];
    const float* b1 = (const float*)d_in[2];
    const float* w2 = (const float*)d_in[3];
    const float* b2 = (const float*)d_in[4];
    float* out = (float*)d_out;

    int rows = in_sizes[0] / NFEAT;              // 131072
    dim3 grid(rows / ROWS_PER_BLOCK);            // 1024 blocks x 256 threads (8 waves)
    net_pool_mlp_kernel<<<grid, 256, SMEM_BYTES, stream>>>(x, w1, b1, w2, b2, out);
}